// GCN_32323923870230
// MI455X (gfx1250) — compile-verified
//
#include <hip/hip_runtime.h>
#include <math.h>

// GCN on SPD manifold: logm -> graph-mix -> expm -> W X W^T -> fro-normalize -> sqrtm.
// All matrix functions expressed as 32x32-padded fp32 GEMM iterations so they run on
// v_wmma_f32_16x16x4_f32 (one wave per matrix, fragments staged through LDS).

typedef float v2f __attribute__((ext_vector_type(2)));
typedef float v8f __attribute__((ext_vector_type(8)));

#define NB 16
#define MB 3
#define TB 32
#define VB 25
#define PB 24
#define TOT (NB * MB * TB * VB) /* 38400 matrices */

__device__ __forceinline__ float wave_reduce_sum(float s) {
#pragma unroll
  for (int off = 16; off > 0; off >>= 1) s += __shfl_xor(s, off, 32);
  return s;
}

// D = alpha*(A @ B) + beta*I for 32x32 row-major (stride 32) LDS matrices.
// Executed by ONE full wave (EXEC all ones). 4 C-tiles x 8 K-steps = 32 WMMAs.
// Fragment layouts per CDNA5 ISA 7.12.2:
//   A 16x4 f32 : lane owns row m; K = {0,1} (lanes 0-15) / {2,3} (lanes 16-31) in 2 VGPRs
//   B 4x16 f32 : lane owns col n; same K half-split across the 2 VGPRs
//   C/D 16x16  : VGPR r -> M = r (lanes 0-15) / r+8 (lanes 16-31), N = lane&15
__device__ __forceinline__ void wave_mm32(float* D, const float* A, const float* B,
                                          float alpha, float beta) {
  const int lane = threadIdx.x & 31;
  const int lh = lane >> 4;
  const int ll = lane & 15;
#pragma unroll
  for (int ti = 0; ti < 2; ++ti) {
#pragma unroll
    for (int tj = 0; tj < 2; ++tj) {
      v8f acc = {0.f, 0.f, 0.f, 0.f, 0.f, 0.f, 0.f, 0.f};
      const int am = ti * 16 + ll;  // A row owned by this lane
      const int bn = tj * 16 + ll;  // B col owned by this lane
#pragma unroll
      for (int kk = 0; kk < 8; ++kk) {
        const int k0 = kk * 4 + lh * 2;
        v2f af, bf;
        af.x = A[am * 32 + k0];
        af.y = A[am * 32 + k0 + 1];
        bf.x = B[k0 * 32 + bn];
        bf.y = B[(k0 + 1) * 32 + bn];
        acc = __builtin_amdgcn_wmma_f32_16x16x4_f32(false, af, false, bf,
                                                    (short)0, acc, false, false);
      }
#pragma unroll
      for (int r = 0; r < 8; ++r) {
        const int mr = ti * 16 + lh * 8 + r;
        float vv = alpha * acc[r];
        if (mr == bn) vv += beta;
        D[mr * 32 + bn] = vv;
      }
    }
  }
}

__device__ __forceinline__ void wave_set_identity(float* D, float dscale) {
  const int lane = threadIdx.x & 31;
  for (int e = lane; e < 1024; e += 32) {
    int r = e >> 5, c = e & 31;
    D[e] = (r == c) ? dscale : 0.f;
  }
}

// Coupled Newton-Schulz sqrt: y <- y^(1/2) in place (eigs of y must lie in (0,3)).
// z/t/u are scratch. iters MUST be a multiple of 3 so the internal (y,u,z)
// rotation returns every buffer to its original role.
__device__ __forceinline__ void wave_ns_sqrt(float* y, float* z, float* t, float* u,
                                             int iters) {
  wave_set_identity(z, 1.f);
#pragma unroll 1
  for (int i = 0; i < iters; ++i) {
    wave_mm32(t, z, y, -0.5f, 1.5f);  // t = 1.5 I - 0.5 z y
    wave_mm32(u, y, t, 1.f, 0.f);     // u = y t      (Y_new)
    wave_mm32(y, t, z, 1.f, 0.f);     // y = t z      (Z_new; old y dead)
    float* tmp = y;                   // rotate roles: y<-u, u<-z, z<-old y
    y = u; u = z; z = tmp;
  }
}

// ---------------- Kernel 1: per-matrix logm via inverse scaling & squaring ----------------
__global__ __launch_bounds__(64) void k_logm(const float* __restrict__ x,
                                             float* __restrict__ lg) {
  __shared__ float smem[2][4][1024];  // 32 KB: 2 waves x 4 ping-pong 32x32 buffers
  const int wid = threadIdx.x >> 5;
  const int lane = threadIdx.x & 31;
  const int idx = blockIdx.x * 2 + wid;
  if (idx >= TOT) return;

  const float* src = x + (size_t)idx * 576;
  float* b0 = smem[wid][0];
  float* b1 = smem[wid][1];
  float* b2 = smem[wid][2];
  float* b3 = smem[wid][3];

  // c = ||X||_F  (spectral <= Frobenius, so X/c has eigenvalues in (0,1])
  float s = 0.f;
  for (int e = lane; e < 576; e += 32) { float v = src[e]; s += v * v; }
  const float c = sqrtf(wave_reduce_sum(s)) + 1e-20f;
  const float invc = 1.f / c;
  const float lnc = logf(c);

  // b0 = X/c embedded in 32x32, pad block = I (log of pad -> 0)
  for (int e = lane; e < 1024; e += 32) {
    int r = e >> 5, cc = e & 31;
    b0[e] = (r < 24 && cc < 24) ? src[r * 24 + cc] * invc
                                : ((r == cc) ? 1.f : 0.f);
  }

  // 4 square-root stages: b0 <- (X/c)^(1/16); eigenvalues -> [~0.84, 1]
  wave_ns_sqrt(b0, b1, b2, b3, 12);
  wave_ns_sqrt(b0, b1, b2, b3, 9);
  wave_ns_sqrt(b0, b1, b2, b3, 9);
  wave_ns_sqrt(b0, b1, b2, b3, 9);

  // E = b0 - I (into b1; pad of E = 0)
  for (int e = lane; e < 1024; e += 32) {
    int r = e >> 5, cc = e & 31;
    b1[e] = b0[e] - ((r == cc) ? 1.f : 0.f);
  }
  // log(I+E) = E*(a1 I + E*(a2 I + ... a7 I)),  a_k = (-1)^(k+1)/k  (Horner)
  float* g = b2;
  float* gn = b3;
  wave_set_identity(g, 1.f / 7.f);
#pragma unroll 1
  for (int k = 6; k >= 1; --k) {
    const float ak = ((k & 1) ? 1.f : -1.f) / (float)k;
    wave_mm32(gn, b1, g, 1.f, ak);  // G = a_k I + E*G
    float* tmp = g; g = gn; gn = tmp;
  }
  // log(X/c) = 2^4 * E * G ; b0 (=X^(1/16)) is dead, use as destination
  wave_mm32(b0, b1, g, 16.f, 0.f);

  float* dst = lg + (size_t)idx * 576;
  for (int e = lane; e < 576; e += 32) {
    int r = e / 24, cc = e - r * 24;
    dst[e] = b0[r * 32 + cc] + ((r == cc) ? lnc : 0.f);  // + ln(c)*I
  }
}

// ------- Kernel 2: aggregate -> expm -> W X W^T -> normalize -> sqrtm -------
__global__ __launch_bounds__(64) void k_gcn_tail(const float* __restrict__ lg,
                                                 const float* __restrict__ A,
                                                 const float* __restrict__ W,
                                                 float* __restrict__ out) {
  __shared__ float smem[2][4][1024];
  __shared__ float Wl[1024], WTl[1024];  // W and W^T, zero-padded to 32x32
  const int wid = threadIdx.x >> 5;
  const int lane = threadIdx.x & 31;

  for (int e = threadIdx.x; e < 1024; e += blockDim.x) {
    int r = e >> 5, cc = e & 31;
    const bool in = (r < 24 && cc < 24);
    Wl[e]  = in ? W[r * 24 + cc] : 0.f;
    WTl[e] = in ? W[cc * 24 + r] : 0.f;
  }
  __syncthreads();

  const int idx = blockIdx.x * 2 + wid;  // flat (n,m,t,j)
  if (idx >= TOT) return;
  const int j = idx % VB;
  const int nmt = idx / VB;

  float* b0 = smem[wid][0];
  float* b1 = smem[wid][1];
  float* b2 = smem[wid][2];
  float* b3 = smem[wid][3];

  // tangent-space aggregation: b0 = sum_v A[nmt,v,j] * lg[nmt,v]   (pad = 0)
  const float* Ab = A + (size_t)nmt * VB * VB;
  const float* Lb = lg + (size_t)nmt * VB * 576;
  for (int e = lane; e < 1024; e += 32) {
    int r = e >> 5, cc = e & 31;
    float sacc = 0.f;
    if (r < 24 && cc < 24) {
      const int off = r * 24 + cc;
#pragma unroll 1
      for (int v = 0; v < VB; ++v) sacc += Ab[v * VB + j] * Lb[(size_t)v * 576 + off];
    }
    b0[e] = sacc;
  }

  // expm via scaling & squaring: pick s so ||Y/2^s||_F <= 0.25, Taylor deg 8
  float s = 0.f;
  for (int e = lane; e < 1024; e += 32) { float v = b0[e]; s += v * v; }
  const float fro = sqrtf(wave_reduce_sum(s));
  int sh = 0;
  if (fro > 0.25f) sh = (int)ceilf(log2f(fro * 4.f));
  if (sh > 14) sh = 14;
  const float sc = exp2f((float)-sh);
  for (int e = lane; e < 1024; e += 32) b0[e] *= sc;

  float* g = b1;
  float* gn = b2;
  wave_set_identity(g, 1.f);  // pad of exp stays I (B pad = 0)
#pragma unroll 1
  for (int k = 8; k >= 1; --k) {
    wave_mm32(gn, b0, g, 1.f / (float)k, 1.f);  // G = I + (B/k)*G
    float* tmp = g; g = gn; gn = tmp;
  }
#pragma unroll 1
  for (int i = 0; i < sh; ++i) {  // repeated squaring
    wave_mm32(gn, g, g, 1.f, 0.f);
    float* tmp = g; g = gn; gn = tmp;
  }

  // congruence: Z = W * expm * W^T  (zero-padded W kills the identity pad)
  float* t1 = gn;  // the non-g buffer of {b1,b2}
  wave_mm32(t1, Wl, g, 1.f, 0.f);
  wave_mm32(b0, t1, WTl, 1.f, 0.f);  // b0 dead -> destination; pad block = 0

  // Frobenius normalization (pad is exactly zero, so full-buffer norm is correct)
  s = 0.f;
  for (int e = lane; e < 1024; e += 32) { float v = b0[e]; s += v * v; }
  const float nf = sqrtf(wave_reduce_sum(s)) + 1e-20f;
  const float innf = 1.f / nf;
  for (int e = lane; e < 1024; e += 32) b0[e] *= innf;

  // sym_powm(., 0.5) == matrix sqrt; eigenvalues in [0,1] -> Newton-Schulz
  wave_ns_sqrt(b0, g, t1, b3, 15);

  float* dst = out + (size_t)idx * 576;
  for (int e = lane; e < 576; e += 32) {
    int r = e / 24, cc = e - r * 24;
    dst[e] = b0[r * 32 + cc];
  }
}

extern "C" void kernel_launch(void* const* d_in, const int* in_sizes, int n_in,
                              void* d_out, int out_size, void* d_ws, size_t ws_size,
                              hipStream_t stream) {
  (void)in_sizes; (void)n_in; (void)out_size; (void)ws_size;
  const float* x = (const float*)d_in[0];  // (16,3,32,25,24,24)
  const float* A = (const float*)d_in[1];  // (16,3,32,25,25)
  const float* W = (const float*)d_in[2];  // (24,24)
  float* out = (float*)d_out;              // (16,3,32,25,24,24)
  float* lg = (float*)d_ws;                // 38400*576 floats of logm scratch

  dim3 blk(64);           // 2 wave32 waves, one 24x24 matrix per wave
  dim3 grd(TOT / 2);      // 19200 blocks, exact
  hipLaunchKernelGGL(k_logm, grd, blk, 0, stream, x, lg);
  hipLaunchKernelGGL(k_gcn_tail, grd, blk, 0, stream, lg, A, W, out);
}